// DNS_31671088841216
// MI455X (gfx1250) — compile-verified
//
#include <hip/hip_runtime.h>
#include <math.h>
#include <stdint.h>

// ---- problem constants (match reference) ----
#define BATCH   4096
#define K_NEGS  64
#define EMBED   128
#define REGS    1e-5f
#define WAVES_PER_BLOCK 2   // 64 threads/block = 2 wave32

typedef __attribute__((ext_vector_type(2))) float v2f;
typedef __attribute__((ext_vector_type(4))) float v4f;
typedef __attribute__((ext_vector_type(8))) float v8f;

__device__ __forceinline__ float wave_reduce_sum(float v) {
    #pragma unroll
    for (int off = 16; off > 0; off >>= 1)
        v += __shfl_xor(v, off, 32);
    return v;
}

// One 512B embedding row per instruction: lane L copies 16B from
// global (row + L*16) straight into LDS (buf + L*16). Tracked by ASYNCcnt.
__device__ __forceinline__ void async_row_to_lds(const float* gsrc, float* ldst) {
    uint32_t laddr = (uint32_t)(uintptr_t)ldst;   // flat->LDS: low 32 bits
    asm volatile("global_load_async_to_lds_b128 %0, %1, off"
                 :: "v"(laddr), "v"(gsrc)
                 : "memory");
}

// Issue one tile's 16 async row copies from pre-loaded indices (VALU-only
// address math between the asm issues: no loads serialized by the clobbers).
__device__ __forceinline__ void issue_tile_async(
    int4 q0, int4 q1, int4 q2, int4 q3,
    const float* __restrict__ item_emb, float* buf, int lane)
{
    const int idx[16] = {q0.x, q0.y, q0.z, q0.w,  q1.x, q1.y, q1.z, q1.w,
                         q2.x, q2.y, q2.z, q2.w,  q3.x, q3.y, q3.z, q3.w};
    #pragma unroll
    for (int r = 0; r < 16; ++r)
        async_row_to_lds(item_emb + (size_t)idx[r] * EMBED + lane * 4,
                         buf + r * EMBED + lane * 4);
}

// One wave per batch element b.
// Phase 1: rank 64 negatives via V_WMMA_F32_16X16X4_F32 (exact f32 math).
//          Tiles staged with async global->LDS DMA, double buffered; neg
//          indices are pipelined one stage deeper so neither the index
//          fetch nor the row gather latency sits on the critical path.
// Phase 2: argmax, BPR loss term and L2 reg term -> per-b scratch slots.
__global__ __launch_bounds__(64) void bpr_rank_kernel(
    const int* __restrict__ user,
    const int* __restrict__ pos,
    const int* __restrict__ negs,
    const float* __restrict__ user_emb,
    const float* __restrict__ item_emb,
    float* __restrict__ ws)
{
    __shared__ float ldsA[WAVES_PER_BLOCK][2][16 * EMBED]; // 2 x 8KB tile buffers / wave
    __shared__ float ldsU[WAVES_PER_BLOCK][EMBED];         // staged user row / wave

    const int lane = threadIdx.x & 31;
    const int wid  = threadIdx.x >> 5;
    const int b    = blockIdx.x * WAVES_PER_BLOCK + wid;   // grid sized exactly

    const int half = lane >> 4;    // 0: lanes 0-15, 1: lanes 16-31
    const int mrow = lane & 15;    // neg row within tile (A matrix M index)

    float* U = &ldsU[wid][0];
    const int* ntab = negs + b * K_NEGS;

    // ---- prologue: tile 0 indices (one clause, one wait), launch tile 0 ----
    {
        int4 q0 = *(const int4*)(ntab + 0);
        int4 q1 = *(const int4*)(ntab + 4);
        int4 q2 = *(const int4*)(ntab + 8);
        int4 q3 = *(const int4*)(ntab + 12);
        issue_tile_async(q0, q1, q2, q3, item_emb, &ldsA[wid][0][0], lane);
    }
    // indices for tile 1, in flight while tile 0 rows stream in
    int4 i0 = *(const int4*)(ntab + 16);
    int4 i1 = *(const int4*)(ntab + 20);
    int4 i2 = *(const int4*)(ntab + 24);
    int4 i3 = *(const int4*)(ntab + 28);

    // stage user row: coalesced b128 load -> LDS (kept in regs for epilogue)
    const float* urow = user_emb + (size_t)user[b] * EMBED;
    v4f uv = *(const v4f*)(urow + lane * 4);
    *(v4f*)(U + lane * 4) = uv;

    // hoist the pos row load: index known now, hide its latency under ranking
    v4f pv = *(const v4f*)(item_emb + (size_t)pos[b] * EMBED + lane * 4);

    float bestVal = -INFINITY;
    int   bestK   = 0;

    #pragma unroll
    for (int t = 0; t < 4; ++t) {
        float* A = &ldsA[wid][t & 1][0];

        if (t < 3) {
            // issue next tile's asyncs from already-loaded indices (VALU only)
            issue_tile_async(i0, i1, i2, i3, item_emb,
                             &ldsA[wid][(t + 1) & 1][0], lane);
            // start loading tile t+2's indices; a whole WMMA tile to complete
            if (t < 2) {
                const int* p = ntab + (t + 2) * 16;
                i0 = *(const int4*)(p + 0);
                i1 = *(const int4*)(p + 4);
                i2 = *(const int4*)(p + 8);
                i3 = *(const int4*)(p + 12);
            }
            // in-order completion: <=16 outstanding means tile t has landed
            asm volatile("s_wait_asynccnt 16" ::: "memory");
        } else {
            asm volatile("s_wait_asynccnt 0" ::: "memory");
        }

        // ---- 16 dots x D=128: two independent WMMA chains (no D->C stalls) ----
        v8f acc0 = {0.f, 0.f, 0.f, 0.f, 0.f, 0.f, 0.f, 0.f};
        v8f acc1 = {0.f, 0.f, 0.f, 0.f, 0.f, 0.f, 0.f, 0.f};
        #pragma unroll 4
        for (int s = 0; s < 16; ++s) {
            // A 16x4 f32 layout: lanes0-15 hold K=0,1 ; lanes16-31 hold K=2,3
            v2f a0 = *(const v2f*)(A + mrow * EMBED + s * 4 + half * 2);
            v2f b0 = *(const v2f*)(U + s * 4 + half * 2);
            v2f a1 = *(const v2f*)(A + mrow * EMBED + (s + 16) * 4 + half * 2);
            v2f b1 = *(const v2f*)(U + (s + 16) * 4 + half * 2);
            acc0 = __builtin_amdgcn_wmma_f32_16x16x4_f32(
                false, a0, false, b0, (short)0, acc0, false, false);
            acc1 = __builtin_amdgcn_wmma_f32_16x16x4_f32(
                false, a1, false, b1, (short)0, acc1, false, false);
        }

        // D layout: VGPR j -> M = j + 8*half (values replicated across N columns)
        #pragma unroll
        for (int j = 0; j < 8; ++j) {
            float v = acc0[j] + acc1[j];
            int   k = t * 16 + half * 8 + j;
            if (v > bestVal) { bestVal = v; bestK = k; }   // strict > = first-occurrence
        }
    }

    // cross-half argmax combine (values identical within each 16-lane half)
    {
        float ov = __shfl_xor(bestVal, 16, 32);
        int   ok = __shfl_xor(bestK,   16, 32);
        if (ov > bestVal || (ov == bestVal && ok < bestK)) { bestVal = ov; bestK = ok; }
    }

    // ---- epilogue: scores + L2 terms (selected row hot in cache) ----
    const int gneg = ntab[bestK];                                // uniform
    v4f nv = *(const v4f*)(item_emb + (size_t)gneg * EMBED + lane * 4);

    float su2 = 0.f, sp2 = 0.f, sn2 = 0.f, sup = 0.f, sun = 0.f;
    #pragma unroll
    for (int j = 0; j < 4; ++j) {
        float u = uv[j], p = pv[j], n = nv[j];
        su2 += u * u;  sp2 += p * p;  sn2 += n * n;
        sup += u * p;  sun += u * n;
    }
    su2 = wave_reduce_sum(su2);
    sp2 = wave_reduce_sum(sp2);
    sn2 = wave_reduce_sum(sn2);
    sup = wave_reduce_sum(sup);
    sun = wave_reduce_sum(sun);

    if (lane == 0) {
        // -log_sigmoid(pos - neg) = softplus(neg - pos), numerically stable
        float x  = sun - sup;
        float lb = fmaxf(x, 0.f) + log1pf(expf(-fabsf(x)));
        ws[b]         = lb;
        ws[BATCH + b] = 0.5f * (su2 + sp2 + sn2);
    }
}

// Deterministic final reduction -> (loss, reg_loss)
__global__ __launch_bounds__(256) void bpr_finalize_kernel(
    const float* __restrict__ ws, float* __restrict__ out)
{
    __shared__ float sl[256];
    __shared__ float sr[256];
    const int t = threadIdx.x;
    float l = 0.f, r = 0.f;
    for (int i = t; i < BATCH; i += 256) {
        l += ws[i];
        r += ws[BATCH + i];
    }
    sl[t] = l; sr[t] = r;
    __syncthreads();
    #pragma unroll
    for (int s = 128; s > 0; s >>= 1) {
        if (t < s) { sl[t] += sl[t + s]; sr[t] += sr[t + s]; }
        __syncthreads();
    }
    if (t == 0) {
        out[0] = sl[0] / (float)BATCH;
        out[1] = REGS * (sr[0] / (float)BATCH);
    }
}

extern "C" void kernel_launch(void* const* d_in, const int* in_sizes, int n_in,
                              void* d_out, int out_size, void* d_ws, size_t ws_size,
                              hipStream_t stream) {
    (void)in_sizes; (void)n_in; (void)out_size; (void)ws_size;
    const int*   user     = (const int*)  d_in[0];
    const int*   pos      = (const int*)  d_in[1];
    const int*   negs     = (const int*)  d_in[2];
    const float* user_emb = (const float*)d_in[3];
    const float* item_emb = (const float*)d_in[4];
    float*       out      = (float*)d_out;
    float*       ws       = (float*)d_ws;   // needs 2*BATCH floats = 32 KB

    dim3 grid(BATCH / WAVES_PER_BLOCK);     // 2048 blocks
    dim3 block(32 * WAVES_PER_BLOCK);       // 64 threads = 2 wave32
    bpr_rank_kernel<<<grid, block, 0, stream>>>(user, pos, negs, user_emb, item_emb, ws);
    bpr_finalize_kernel<<<1, 256, 0, stream>>>(ws, out);
}